// FAPELoss_89292370084240
// MI455X (gfx1250) — compile-verified
//
#include <hip/hip_runtime.h>
#include <hip/hip_bf16.h>

typedef __attribute__((ext_vector_type(2))) float v2f;
typedef __attribute__((ext_vector_type(8))) float v8f;

#define AS1 __attribute__((address_space(1)))
#define AS3 __attribute__((address_space(3)))

#define EPSF        1e-8f
#define CLAMP_DIST  10.0f
#define CLASH_W     0.05f
#define PHYS_W      0.3f
#define PTILE       128   // atoms per pair-tile side (8 waves x 16 rows)

#if __has_builtin(__builtin_amdgcn_global_load_async_to_lds_b32)
#define HAVE_ASYNC_LDS 1
#else
#define HAVE_ASYNC_LDS 0
#endif

__device__ __forceinline__ void wait_asynccnt0() {
#if __has_builtin(__builtin_amdgcn_s_wait_asynccnt)
    __builtin_amdgcn_s_wait_asynccnt(0);
#else
    asm volatile("s_wait_asynccnt 0" ::: "memory");
#endif
}

// -------- workspace layout (first 16 dwords of d_ws) --------
// f32 view: [0] fape_sum, [3+b] physics penalty sum per batch
// u32 view: [1+b] clash count per batch, [5+b] C-N pair count per batch
__global__ void init_ws_kernel(unsigned int* ws) {
    if (threadIdx.x < 16) ws[threadIdx.x] = 0u;
}

// ======================= FAPE =======================
// one wave per frame; lane-per-atom loop; everything L2-resident.
__global__ void fape_kernel(const float* __restrict__ rots_pred,
                            const float* __restrict__ trans_pred,
                            const float* __restrict__ coords_pred,
                            const float* __restrict__ rots_true,
                            const float* __restrict__ trans_true,
                            const float* __restrict__ coords_true,
                            const float* __restrict__ res_mask,
                            const float* __restrict__ mask,
                            float* __restrict__ ws_f,
                            int B, int R, int A) {
    const int wave  = threadIdx.x >> 5;
    const int lane  = threadIdx.x & 31;
    const int frame = blockIdx.x * (blockDim.x >> 5) + wave;
    if (frame >= B * R) return;
    const int b = frame / R;

    const float* Rp = rots_pred + (size_t)frame * 9;
    const float* Rt = rots_true + (size_t)frame * 9;
    const float* tp = trans_pred + (size_t)frame * 3;
    const float* tt = trans_true + (size_t)frame * 3;

    float rp[9], rt[9];
#pragma unroll
    for (int i = 0; i < 9; ++i) { rp[i] = Rp[i]; rt[i] = Rt[i]; }
    const float tpx = tp[0], tpy = tp[1], tpz = tp[2];
    const float ttx = tt[0], tty = tt[1], ttz = tt[2];
    const float rmask = res_mask[frame];

    const float* Xp = coords_pred + (size_t)b * A * 3;
    const float* Xt = coords_true + (size_t)b * A * 3;
    const float* Mk = mask + (size_t)b * A;

    float acc = 0.0f;
    for (int a = lane; a < A; a += 32) {
        // lp = R^T (x - t)   (invert_frames + apply)
        float xp0 = Xp[a*3+0] - tpx, xp1 = Xp[a*3+1] - tpy, xp2 = Xp[a*3+2] - tpz;
        float xt0 = Xt[a*3+0] - ttx, xt1 = Xt[a*3+1] - tty, xt2 = Xt[a*3+2] - ttz;
        float lp0 = rp[0]*xp0 + rp[3]*xp1 + rp[6]*xp2;
        float lp1 = rp[1]*xp0 + rp[4]*xp1 + rp[7]*xp2;
        float lp2 = rp[2]*xp0 + rp[5]*xp1 + rp[8]*xp2;
        float lt0 = rt[0]*xt0 + rt[3]*xt1 + rt[6]*xt2;
        float lt1 = rt[1]*xt0 + rt[4]*xt1 + rt[7]*xt2;
        float lt2 = rt[2]*xt0 + rt[5]*xt1 + rt[8]*xt2;
        float d0 = lp0 - lt0, d1 = lp1 - lt1, d2 = lp2 - lt2;
        float err = sqrtf(d0*d0 + d1*d1 + d2*d2 + EPSF);
        err = fminf(err, CLAMP_DIST);
        acc += err * Mk[a] * rmask;
    }
#pragma unroll
    for (int off = 16; off > 0; off >>= 1) acc += __shfl_xor(acc, off, 32);
    if (lane == 0) unsafeAtomicAdd(&ws_f[0], acc);
}

// ================= pairwise clash + physics =================
// 128x128 atom tile per block; Gram cross-term via V_WMMA_F32_16X16X4_F32
// (K=3 coords zero-padded to K=4).
// A/B f32 fragments (16x4, 2 VGPRs), per ISA layout:
//   lanes 0-15 : {K0,K1} of row (lane)    lanes 16-31: {K2,K3=0} of row (lane-16)
// -> with LDS records {x,y,z,0}, each fragment is ONE ds_load_b64 at
//    record_base + (lane>=16 ? 8 : 0).
// C/D layout: VGPR v -> M = v + 8*(lane>=16), N = lane & 15.
// ct loop capped at unroll 2: one v8f accumulator pair live at a time
// (full unroll caused scratch spills); 8 waves/block hide WMMA latency.
__global__ void pair_kernel(const float* __restrict__ coords_pred,
                            const int*   __restrict__ atom_types,
                            const float* __restrict__ vdw_radii,
                            float* __restrict__ ws_f,
                            unsigned int* __restrict__ ws_u,
                            int A) {
    __shared__ __align__(16) float rX[PTILE * 4];   // {x,y,z,0} per row atom
    __shared__ __align__(16) float cX[PTILE * 4];   // {x,y,z,0} per col atom
    __shared__ float rSq[PTILE],  cSq[PTILE];
    __shared__ float rRad[PTILE], cRad[PTILE];
    __shared__ int   rTy[PTILE],  cTy[PTILE];

    const int b       = blockIdx.z;
    const int rowBase = blockIdx.y * PTILE;
    const int colBase = blockIdx.x * PTILE;
    const float* X = coords_pred + (size_t)b * A * 3;
    const int*   T = atom_types  + (size_t)b * A;

    // ---- phase 1: stage coords global -> LDS (async DMA, ASYNCcnt) ----
    const int t     = threadIdx.x;
    const int isCol = t >= PTILE;
    const int idx   = isCol ? (t - PTILE) : t;
    const int atom  = (isCol ? colBase : rowBase) + idx;
    float* dX = isCol ? cX : rX;

#if HAVE_ASYNC_LDS
#pragma unroll
    for (int c = 0; c < 3; ++c) {
        __builtin_amdgcn_global_load_async_to_lds_b32(
            (AS1 int*)(X + (size_t)atom * 3 + c),
            (AS3 int*)(dX + idx * 4 + c),
            /*offset=*/0, /*cpol=*/0);
    }
#else
    {
        float x = X[atom*3+0], y = X[atom*3+1], z = X[atom*3+2];
        dX[idx*4+0] = x; dX[idx*4+1] = y; dX[idx*4+2] = z;
    }
#endif
    // types + radii via normal path (value needed for the gather anyway)
    {
        int ty = T[atom];
        (isCol ? cRad : rRad)[idx] = vdw_radii[ty];
        (isCol ? cTy  : rTy )[idx] = ty;
        dX[idx*4+3] = 0.0f;          // K=3 zero pad for the WMMA fragment
    }
#if HAVE_ASYNC_LDS
    wait_asynccnt0();
#endif
    __syncthreads();

    // ---- phase 2: squared norms from staged coords ----
    {
        float x = dX[idx*4+0], y = dX[idx*4+1], z = dX[idx*4+2];
        (isCol ? cSq : rSq)[idx] = x*x + y*y + z*z;
    }
    __syncthreads();

    const int lane = t & 31;
    const int wave = t >> 5;        // wave w owns rows [16w, 16w+16)
    const int hi   = lane >> 4;
    const int li   = lane & 15;

    // A fragment: one ds_load_b64, constant over the ct loop
    const v2f afrag = *(const v2f*)(rX + (wave * 16 + li) * 4 + hi * 2);

    // hoist per-lane row-side epilogue data (M = wave*16 + v + 8*hi)
    float mSq[8], mRad[8]; int mIsC[8];
#pragma unroll
    for (int v = 0; v < 8; ++v) {
        int mm  = wave * 16 + v + 8 * hi;
        mSq[v]  = rSq[mm];
        mRad[v] = rRad[mm];
        mIsC[v] = (rTy[mm] == 0);           // C_IDX
    }

    int   clashCnt = 0, cnCnt = 0;
    float pen = 0.0f;

#pragma unroll 2
    for (int ct = 0; ct < PTILE / 16; ++ct) {
        const int nb = ct * 16 + li;
        const v2f bfrag = *(const v2f*)(cX + nb * 4 + hi * 2);

        v8f cacc = {};
        v8f g = __builtin_amdgcn_wmma_f32_16x16x4_f32(
            /*neg_a=*/false, afrag, /*neg_b=*/false, bfrag,
            /*c_mod=*/(short)0, cacc, /*reuse_a=*/false, /*reuse_b=*/false);

        const float nSq  = cSq[nb];
        const float nRad = cRad[nb];
        const int   nIsN = (cTy[nb] == 1);  // N_IDX
#pragma unroll
        for (int v = 0; v < 8; ++v) {
            float d2   = fmaf(-2.0f, g[v], mSq[v] + nSq);
            float dist = sqrtf(fmaxf(d2, 1e-12f));
            float mind = mRad[v] + nRad;
            clashCnt += ((dist < mind) & (dist > EPSF)) ? 1 : 0;
            int cn = mIsC[v] & nIsN;
            cnCnt += cn;
            pen   += cn ? fmaxf(fabsf(dist - 1.33f) - 0.2f, 0.0f) : 0.0f;
        }
    }

#pragma unroll
    for (int off = 16; off > 0; off >>= 1) {
        clashCnt += __shfl_xor(clashCnt, off, 32);
        cnCnt    += __shfl_xor(cnCnt,    off, 32);
        pen      += __shfl_xor(pen,      off, 32);
    }
    if (lane == 0) {
        atomicAdd(&ws_u[1 + b], (unsigned int)clashCnt);
        atomicAdd(&ws_u[5 + b], (unsigned int)cnCnt);
        unsafeAtomicAdd(&ws_f[3 + b], pen);
    }
}

// ======================= finalize =======================
__global__ void finalize_kernel(const float* __restrict__ res_mask,
                                const float* __restrict__ mask,
                                const float* __restrict__ ws_f,
                                const unsigned int* __restrict__ ws_u,
                                float* __restrict__ out,
                                int B, int R, int A) {
    __shared__ float sred[256];
    const int t = threadIdx.x;

    float ms = 0.0f;
    for (int i = t; i < B * A; i += blockDim.x) ms += mask[i];
    sred[t] = ms; __syncthreads();
    for (int s = 128; s > 0; s >>= 1) { if (t < s) sred[t] += sred[t + s]; __syncthreads(); }
    const float maskSum = sred[0]; __syncthreads();

    float rs = 0.0f;
    for (int i = t; i < B * R; i += blockDim.x) rs += res_mask[i];
    sred[t] = rs; __syncthreads();
    for (int s = 128; s > 0; s >>= 1) { if (t < s) sred[t] += sred[t + s]; __syncthreads(); }
    const float resSum = sred[0];

    if (t == 0) {
        float fape  = ws_f[0] / (maskSum * resSum + EPSF);
        float clash = 0.0f, phys = 0.0f;
        for (int b = 0; b < B; ++b) {
            clash += (0.5f * (float)ws_u[1 + b]) / (float)A;
            phys  += ws_f[3 + b] / fmaxf((float)ws_u[5 + b], 1.0f);
        }
        clash /= (float)B;
        phys  /= (float)B;
        out[0] = fape + CLASH_W * clash + PHYS_W * phys;
    }
}

extern "C" void kernel_launch(void* const* d_in, const int* in_sizes, int n_in,
                              void* d_out, int out_size, void* d_ws, size_t ws_size,
                              hipStream_t stream) {
    const float* rots_pred   = (const float*)d_in[0];
    const float* trans_pred  = (const float*)d_in[1];
    const float* coords_pred = (const float*)d_in[2];
    const float* rots_true   = (const float*)d_in[3];
    const float* trans_true  = (const float*)d_in[4];
    const float* coords_true = (const float*)d_in[5];
    const int*   atom_types  = (const int*)d_in[6];
    const float* vdw_radii   = (const float*)d_in[7];
    const float* res_mask    = (const float*)d_in[8];
    const float* mask        = (const float*)d_in[9];

    const int B = 2;
    const int R = in_sizes[8] / B;   // res_mask is [B,R]
    const int A = in_sizes[9] / B;   // mask is [B,A]

    float*        ws_f = (float*)d_ws;
    unsigned int* ws_u = (unsigned int*)d_ws;

    init_ws_kernel<<<1, 32, 0, stream>>>(ws_u);

    const int frames = B * R;
    fape_kernel<<<(frames + 7) / 8, 256, 0, stream>>>(
        rots_pred, trans_pred, coords_pred,
        rots_true, trans_true, coords_true,
        res_mask, mask, ws_f, B, R, A);

    dim3 pgrid(A / PTILE, A / PTILE, B);
    pair_kernel<<<pgrid, 256, 0, stream>>>(
        coords_pred, atom_types, vdw_radii, ws_f, ws_u, A);

    finalize_kernel<<<1, 256, 0, stream>>>(
        res_mask, mask, ws_f, ws_u, (float*)d_out, B, R, A);
}